// DualReLU_62637803045540
// MI455X (gfx1250) — compile-verified
//
#include <hip/hip_runtime.h>
#include <stdint.h>

typedef int i32x4 __attribute__((ext_vector_type(4)));
typedef int i32x8 __attribute__((ext_vector_type(8)));

#define BATCH 32  // B from the reference (d_next is [B, N])

// One workgroup per batch b. batch_idx is sorted, so rows for batch b form a
// contiguous segment [lo, hi) found by binary search. The 8 KB d_next row for
// this batch is DMA'd into LDS by the Tensor Data Mover while the zl/zu LDS
// accumulators are zeroed and the segment bounds are searched; the scatter
// then runs entirely on LDS float atomics, and each block writes its two
// output planes once with b128 stores.
__global__ __launch_bounds__(256)
void dual_relu_scatter(const int* __restrict__ batch_idx,
                       const int* __restrict__ neuron_idx,
                       const float* __restrict__ d_vals,
                       const float* __restrict__ d_next,
                       const float* __restrict__ zlI,
                       float* __restrict__ out,   // [2, B, N]
                       int K, int N)
{
    extern __shared__ float smem[];       // 3*N floats (N % 4 == 0)
    float* s_dnext = smem;                // [N]  d_next row for this batch
    float* s_zl    = smem + N;            // [N]  zl accumulator
    float* s_zu    = smem + 2 * N;        // [N]  zu accumulator

    const int b    = blockIdx.x;
    const int tid  = threadIdx.x;
    const int nthr = blockDim.x;

    // ---- TDM async tensor load: d_next[b, 0:N] -> s_dnext (wave 0 issues) ----
    // Overlaps with accumulator zeroing + segment binary search below.
    if (tid < 32) {
        uint64_t gaddr   = (uint64_t)(uintptr_t)(const void*)(d_next + (size_t)b * (size_t)N);
        // Flat LDS address low 32 bits == LDS byte offset (aperture mapping).
        int lds_off = __builtin_amdgcn_readfirstlane((int)(uint32_t)(uintptr_t)(void*)s_dnext);
        int ga_lo   = __builtin_amdgcn_readfirstlane((int)(uint32_t)gaddr);
        int ga_hi   = __builtin_amdgcn_readfirstlane((int)(uint32_t)((gaddr >> 32) & 0x1FFFFFFu));
        uint32_t dim0 = (uint32_t)N;

        // D# group 0 (128b): count=1 | lds_addr | global_addr[56:0] | type=2
        i32x4 g0;
        g0.x = 1;                                             // count=1, user descriptor
        g0.y = lds_off;                                       // lds_addr
        g0.z = ga_lo;                                         // global_addr[31:0]
        g0.w = ga_hi | (int)0x80000000u;                      // global_addr[56:32] | type=2

        // D# group 1 (256b): data_size=4B, tensor_dim0=N, tensor_dim1=1,
        // tile_dim0=N, tile_dim1=1, stride0=stride1=N
        i32x8 g1;
        g1[0] = (int)(2u << 16);                               // data_size=4B; no multicast/pad/iterate
        g1[1] = (int)((dim0 & 0xFFFFu) << 16);                 // tensor_dim0[15:0] @ bits[63:48]
        g1[2] = (int)(((dim0 >> 16) & 0xFFFFu) | (1u << 16));  // tensor_dim0[31:16] | tensor_dim1=1 lo
        g1[3] = (int)((dim0 & 0xFFFFu) << 16);                 // tensor_dim1 hi=0 | tile_dim0=N
        g1[4] = 1;                                             // tile_dim1=1, tile_dim2=0
        g1[5] = (int)dim0;                                     // tensor_dim0_stride[31:0]
        g1[6] = (int)((dim0 & 0xFFFFu) << 16);                 // stride0 hi=0 | stride1[15:0]
        g1[7] = (int)((dim0 >> 16) & 0xFFFFu);                 // stride1[47:16]

        i32x4 g2; g2.x = 1; g2.y = 0; g2.z = 0; g2.w = 0;      // tensor_dim2=1, rest unused
        i32x4 g3; g3.x = 0; g3.y = 0; g3.z = 0; g3.w = 0;

        asm volatile("tensor_load_to_lds %0, %1, %2, %3"
                     :
                     : "s"(g0), "s"(g1), "s"(g2), "s"(g3)
                     : "memory");
    }

    // ---- Zero accumulators with b128 LDS stores (overlaps the DMA) ----
    {
        float4  zero4 = make_float4(0.f, 0.f, 0.f, 0.f);
        float4* acc4  = (float4*)(smem + N);                   // 2N floats = N/2 float4s
        for (int i = tid; i < N / 2; i += nthr) acc4[i] = zero4;
    }

    // ---- Segment bounds for batch b (batch_idx sorted ascending) ----
    int lo, hi;
    {
        int l = 0, r = K;
        while (l < r) { int m = (l + r) >> 1; if (batch_idx[m] < b) l = m + 1; else r = m; }
        lo = l;
        r = K;
        while (l < r) { int m = (l + r) >> 1; if (batch_idx[m] < b + 1) l = m + 1; else r = m; }
        hi = l;
    }

    // ---- Warm L2/L0 for the scatter streams while waiting at the barrier ----
    if (lo + tid < hi) {
        __builtin_prefetch(neuron_idx + lo + tid, 0, 0);       // global_prefetch_b8
        __builtin_prefetch(d_vals + lo + tid, 0, 0);
        __builtin_prefetch(zlI + lo + tid, 0, 0);
    }

    // Wave 0 drains its TDM before signaling; barrier publishes LDS to all.
    if (tid < 32) __builtin_amdgcn_s_wait_tensorcnt(0);        // s_wait_tensorcnt 0
    __syncthreads();

    // ---- Scatter: one row per thread per iteration, LDS float atomics ----
    for (int r = lo + tid; r < hi; r += nthr) {
        int   c = neuron_idx[r];
        float v = s_dnext[c] * d_vals[r];
        float z = zlI[r];
        atomicAdd(&s_zl[c],  z * fmaxf(-v, 0.0f));   // ds_add_f32
        atomicAdd(&s_zu[c], -(z * fmaxf(v, 0.0f)));  // ds_add_f32
    }

    __syncthreads();

    // ---- Coalesced b128 write-out of this batch's two output planes ----
    {
        const float4* zl4 = (const float4*)s_zl;
        const float4* zu4 = (const float4*)s_zu;
        float4* o_zl = (float4*)(out + (size_t)b * N);
        float4* o_zu = (float4*)(out + (size_t)BATCH * N + (size_t)b * N);
        for (int i = tid; i < N / 4; i += nthr) {
            o_zl[i] = zl4[i];                                  // ds_load_b128 + global_store_b128
            o_zu[i] = zu4[i];
        }
    }
}

extern "C" void kernel_launch(void* const* d_in, const int* in_sizes, int n_in,
                              void* d_out, int out_size, void* d_ws, size_t ws_size,
                              hipStream_t stream) {
    const int*   batch_idx  = (const int*)d_in[0];
    const int*   neuron_idx = (const int*)d_in[1];
    const float* d_vals     = (const float*)d_in[2];
    const float* d_next     = (const float*)d_in[3];
    const float* zlI        = (const float*)d_in[4];
    float*       out        = (float*)d_out;

    const int K = in_sizes[0];
    const int N = in_sizes[3] / BATCH;          // d_next is [B, N]
    const size_t shmem = (size_t)3 * (size_t)N * sizeof(float);

    dual_relu_scatter<<<BATCH, 256, shmem, stream>>>(batch_idx, neuron_idx,
                                                     d_vals, d_next, zlI,
                                                     out, K, N);
}